// ProjFuser_2336462209623
// MI455X (gfx1250) — compile-verified
//
#include <hip/hip_runtime.h>
#include <math.h>

typedef __attribute__((ext_vector_type(2))) float v2f;
typedef __attribute__((ext_vector_type(8))) float v8f;

#define EPSV 1e-5f
#define CFC 256
#define HFC 32
#define WFC 88
#define TNC 6
#define NPTS 32768   // B*BW*BH*T*C = 128*128*2

// Uniform-lane broadcast via v_readlane_b32 (SGPR result, no LDS traffic).
__device__ __forceinline__ float readlane_f(float v, int srcLane) {
    return __int_as_float(__builtin_amdgcn_readlane(__float_as_int(v), srcLane));
}

// v_rcp_f32 + one Newton-Raphson step (~0.5 ulp). Divisors here are
// well-conditioned (>= 1e-5, or small integers), so this replaces the
// ~12-instruction IEEE division expansion.
__device__ __forceinline__ float fast_rcp(float x) {
    float r = __builtin_amdgcn_rcpf(x);
    r = r * __builtin_fmaf(-x, r, 2.0f);
    return r;
}

// ---------------------------------------------------------------------------
// Kernel 1: build M[n] = lid2img[n] @ ego2lid[n] and K[n] = img2lid[n] @ M[n]
// padded to 8 cameras (cams 6,7 zero) so the WMMA A-operand pad rows are 0.
// ---------------------------------------------------------------------------
__global__ void setup_mats_kernel(const float* __restrict__ lid2img,
                                  const float* __restrict__ ego2lid,
                                  const float* __restrict__ img2lid,
                                  float* __restrict__ wsM,
                                  float* __restrict__ wsK) {
    int t = threadIdx.x;            // 0..127 : 8 cams x 16 entries
    int n = t >> 4;
    int e = t & 15;
    int i = e >> 2;
    int j = e & 3;
    float mv = 0.f, kv = 0.f;
    if (n < TNC) {
        const float* L = lid2img + n * 16;
        const float* E = ego2lid + n * 16;
        const float* G = img2lid + n * 16;
        #pragma unroll
        for (int k = 0; k < 4; ++k) mv += L[i * 4 + k] * E[k * 4 + j];
        #pragma unroll
        for (int k = 0; k < 4; ++k) {
            float mkj = 0.f;
            #pragma unroll
            for (int l = 0; l < 4; ++l) mkj += L[k * 4 + l] * E[l * 4 + j];
            kv += G[i * 4 + k] * mkj;
        }
    }
    wsM[t] = mv;
    wsK[t] = kv;
}

// ---------------------------------------------------------------------------
// Kernel 2: transpose img_feat (6,256,32,88) -> featT (6,32,88,256) so the
// 256-channel bilerp corner fetches are contiguous 1KB coalesced reads.
// ---------------------------------------------------------------------------
__global__ void transpose_feat_kernel(const float* __restrict__ imgFeat,
                                      float* __restrict__ featT) {
    __shared__ float tile[32][33];
    int xt = blockIdx.x % 3;        // x tile (3 * 32 >= 88)
    int ct = blockIdx.x / 3;        // channel tile (8 * 32 = 256)
    int y  = blockIdx.y;            // 0..31
    int n  = blockIdx.z;            // 0..5
    int tx = threadIdx.x;           // 0..31
    int ty = threadIdx.y;           // 0..7
    int x = xt * 32 + tx;
    #pragma unroll
    for (int r = 0; r < 4; ++r) {
        int cl = ty + r * 8;
        int c  = ct * 32 + cl;
        float v = 0.f;
        if (x < WFC) v = imgFeat[((size_t)(n * CFC + c) * HFC + y) * WFC + x];
        tile[cl][tx] = v;
    }
    __syncthreads();
    int c = ct * 32 + tx;
    #pragma unroll
    for (int r = 0; r < 4; ++r) {
        int xl = ty + r * 8;
        int xx = xt * 32 + xl;
        if (xx < WFC)
            featT[((size_t)(n * HFC + y) * WFC + xx) * CFC + c] = tile[tx][xl];
    }
}

// ---------------------------------------------------------------------------
// Kernel 3: WMMA projection (V_WMMA_F32_16X16X4_F32) + wave-cooperative
// bilinear gather + camera fuse + pair average.
// One wave = 16 points = 8 output pixels. Lane i owns channels [8i, 8i+8).
// ---------------------------------------------------------------------------
__global__ void __launch_bounds__(128)
proj_fuse_kernel(const float* __restrict__ points,
                 const float* __restrict__ featT,
                 const float* __restrict__ wsM,
                 const float* __restrict__ wsK,
                 const int* __restrict__ imgH,
                 const int* __restrict__ imgW,
                 float* __restrict__ out) {
    const int lane = threadIdx.x & 31;
    const int gw   = blockIdx.x * 4 + (threadIdx.x >> 5);
    const int base = gw * 16;                 // first point of this wave

    const int j  = lane & 15;
    const int hi = lane >> 4;

    const float su = (float)WFC / (float)imgW[0];   // 88/704 = 0.125
    const float sv = (float)HFC / (float)imgH[0];   // 32/256 = 0.125

    // B operand: pts_h as 4x16 (rows K=0..3 = x,y,z,1).
    // VGPR0: lanes 0-15 row K=0, lanes 16-31 row K=2; VGPR1: rows K=1 / K=3.
    const float* pp = points + (size_t)(base + j) * 3;
    float px = pp[0], py = pp[1], pz = pp[2];
    v2f bmat;
    bmat.x = hi ? pz : px;
    bmat.y = hi ? 1.0f : py;

    // A operands: 16x4 = four stacked 4x4 camera matrices.
    // A(m,k): lane = (k>=2?16:0)+m, vgpr = k%2  ->  k0 = hi?2:0, k1 = hi?3:1.
    const int m  = j;
    const int k0 = hi ? 2 : 0;
    const int k1 = hi ? 3 : 1;
    v2f a1, a2, g1, g2;
    a1.x = wsM[m * 4 + k0];        a1.y = wsM[m * 4 + k1];        // cams 0-3
    a2.x = wsM[(m + 16) * 4 + k0]; a2.y = wsM[(m + 16) * 4 + k1]; // cams 4-7 (6,7=0)
    g1.x = wsK[m * 4 + k0];        g1.y = wsK[m * 4 + k1];
    g2.x = wsK[(m + 16) * 4 + k0]; g2.y = wsK[(m + 16) * 4 + k1];

    v8f zero = {};
    // D[4*cam + i][p] = proj component i of point p under camera cam
    v8f dp0 = __builtin_amdgcn_wmma_f32_16x16x4_f32(false, a1, false, bmat, (short)0, zero, false, false);
    v8f dp1 = __builtin_amdgcn_wmma_f32_16x16x4_f32(false, a2, false, bmat, (short)0, zero, false, false);
    v8f db0 = __builtin_amdgcn_wmma_f32_16x16x4_f32(false, g1, false, bmat, (short)0, zero, false, false);
    v8f db1 = __builtin_amdgcn_wmma_f32_16x16x4_f32(false, g2, false, bmat, (short)0, zero, false, false);

    const int rowStride = WFC * CFC;          // 22528 floats: y -> y+1

    for (int pair = 0; pair < 8; ++pair) {
        float of[8] = {0.f, 0.f, 0.f, 0.f, 0.f, 0.f, 0.f, 0.f};
        float ox = 0.f, oy = 0.f, oz = 0.f;

        for (int half = 0; half < 2; ++half) {
            const int p = pair * 2 + half;
            float fused[8] = {0.f, 0.f, 0.f, 0.f, 0.f, 0.f, 0.f, 0.f};
            float bs0 = 0.f, bs1 = 0.f, bs2 = 0.f, bs3 = 0.f;
            float cnt = 0.f;

            #pragma unroll
            for (int n = 0; n < 6; ++n) {
                // D layout: VGPR r holds rows r (lanes 0-15) and r+8 (lanes 16-31).
                // cams 0,1 -> rows 0..7 @ lane p ; cams 2,3 -> rows 8..15 @ lane p+16
                // cams 4,5 -> second WMMA rows 0..3 / 4..7 @ lane p, but n=5 reads rows 8..11 @ lane p+16 (zero pad).
                const int  src = p + ((n == 2 || n == 3 || n == 5) ? 16 : 0);
                const int  vb  = (n == 1 || n == 3) ? 4 : 0;
                const v8f  dp  = (n >= 4) ? dp1 : dp0;
                const v8f  db  = (n >= 4) ? db1 : db0;

                // Wave-uniform broadcasts: v_readlane_b32 -> SGPR, no LDS pipe.
                float q0    = readlane_f(dp[vb + 0], src);
                float q1    = readlane_f(dp[vb + 1], src);
                float depth = readlane_f(dp[vb + 2], src);

                float rd = fast_rcp(fmaxf(depth, EPSV));   // shared denominator
                float u = q0 * rd * su;
                float v = q1 * rd * sv;
                int valid = (depth > EPSV) && (u >= 0.f) && (u <= (float)(WFC - 1))
                                           && (v >= 0.f) && (v <= (float)(HFC - 1));
                // Condition is uniform; force a scalar branch (no EXEC masking).
                if (__builtin_amdgcn_readfirstlane(valid)) {
                    cnt += 1.f;
                    int x0 = (int)floorf(u);
                    int y0 = (int)floorf(v);
                    x0 = x0 < 0 ? 0 : (x0 > WFC - 2 ? WFC - 2 : x0);
                    y0 = y0 < 0 ? 0 : (y0 > HFC - 2 ? HFC - 2 : y0);
                    float wx = fminf(fmaxf(u - (float)x0, 0.f), 1.f);
                    float wy = fminf(fmaxf(v - (float)y0, 0.f), 1.f);
                    float w00 = (1.f - wy) * (1.f - wx);
                    float w01 = (1.f - wy) * wx;
                    float w10 = wy * (1.f - wx);
                    float w11 = wy * wx;

                    const float* fb = featT + (size_t)((n * HFC + y0) * WFC + x0) * CFC + lane * 8;
                    const float4* c00 = (const float4*)fb;
                    const float4* c01 = (const float4*)(fb + CFC);
                    const float4* c10 = (const float4*)(fb + rowStride);
                    const float4* c11 = (const float4*)(fb + rowStride + CFC);
                    float4 f00a = c00[0], f00b = c00[1];
                    float4 f01a = c01[0], f01b = c01[1];
                    float4 f10a = c10[0], f10b = c10[1];
                    float4 f11a = c11[0], f11b = c11[1];

                    fused[0] += w00 * f00a.x + w01 * f01a.x + w10 * f10a.x + w11 * f11a.x;
                    fused[1] += w00 * f00a.y + w01 * f01a.y + w10 * f10a.y + w11 * f11a.y;
                    fused[2] += w00 * f00a.z + w01 * f01a.z + w10 * f10a.z + w11 * f11a.z;
                    fused[3] += w00 * f00a.w + w01 * f01a.w + w10 * f10a.w + w11 * f11a.w;
                    fused[4] += w00 * f00b.x + w01 * f01b.x + w10 * f10b.x + w11 * f11b.x;
                    fused[5] += w00 * f00b.y + w01 * f01b.y + w10 * f10b.y + w11 * f11b.y;
                    fused[6] += w00 * f00b.z + w01 * f01b.z + w10 * f10b.z + w11 * f11b.z;
                    fused[7] += w00 * f00b.w + w01 * f01b.w + w10 * f10b.w + w11 * f11b.w;

                    bs0 += readlane_f(db[vb + 0], src);
                    bs1 += readlane_f(db[vb + 1], src);
                    bs2 += readlane_f(db[vb + 2], src);
                    bs3 += readlane_f(db[vb + 3], src);
                }
            }

            float rin = fast_rcp(fmaxf(cnt, 1.f));
            float w   = bs3 * rin;
            float den = (fabsf(w) < EPSV) ? 1.f : w;
            float rdn = 0.5f * fast_rcp(den);
            ox += (bs0 * rin) * rdn;
            oy += (bs1 * rin) * rdn;
            oz += (bs2 * rin) * rdn;
            #pragma unroll
            for (int c = 0; c < 8; ++c) of[c] += 0.5f * fused[c] * rin;
        }

        const int pix = (base >> 1) + pair;     // output pixel (w*128+h)
        float* op = out + (size_t)pix * 259;
        if (lane == 0) { op[0] = ox; op[1] = oy; op[2] = oz; }
        float* oc = op + 3 + lane * 8;
        #pragma unroll
        for (int c = 0; c < 8; ++c) oc[c] = of[c];
    }
}

// ---------------------------------------------------------------------------
extern "C" void kernel_launch(void* const* d_in, const int* in_sizes, int n_in,
                              void* d_out, int out_size, void* d_ws, size_t ws_size,
                              hipStream_t stream) {
    (void)in_sizes; (void)n_in; (void)out_size; (void)ws_size;
    const float* points  = (const float*)d_in[0];
    const float* imgFeat = (const float*)d_in[1];
    const float* lid2img = (const float*)d_in[2];
    const float* ego2lid = (const float*)d_in[3];
    const float* img2lid = (const float*)d_in[4];
    const int*   imgH    = (const int*)d_in[5];
    const int*   imgW    = (const int*)d_in[6];
    float* out = (float*)d_out;

    float* wsM   = (float*)d_ws;                         // 8 cams * 16 f
    float* wsK   = wsM + 128;                            // 8 cams * 16 f
    float* featT = (float*)((char*)d_ws + 1024);         // 6*32*88*256 f ~ 17.3 MB

    hipLaunchKernelGGL(setup_mats_kernel, dim3(1), dim3(128), 0, stream,
                       lid2img, ego2lid, img2lid, wsM, wsK);
    hipLaunchKernelGGL(transpose_feat_kernel, dim3(24, 32, 6), dim3(32, 8), 0, stream,
                       imgFeat, featT);
    // 32768 points / 16 per wave = 2048 waves = 512 blocks of 4 waves
    hipLaunchKernelGGL(proj_fuse_kernel, dim3(512), dim3(128), 0, stream,
                       points, featT, wsM, wsK, imgH, imgW, out);
}